// S3RecModel_75539884802440
// MI455X (gfx1250) — compile-verified
//
#include <hip/hip_runtime.h>

typedef __attribute__((ext_vector_type(2))) float v2f;
typedef __attribute__((ext_vector_type(8))) float v8f;

namespace {
constexpr int kB = 1024;   // batch
constexpr int kS = 200;    // seq length
constexpr int kD = 64;     // embedding dim
constexpr int kC = 256;    // codebook size
constexpr int kXStride = 72;  // padded LDS row stride (floats) to dodge bank conflicts
}

__global__ __launch_bounds__(256) void s3rec_cluster_mean_kernel(
    const float* __restrict__ x,       // [B,S,D] f32
    const int*   __restrict__ labels,  // [B,S]
    const int*   __restrict__ amask,   // [B,S]
    float*       __restrict__ out_emb, // [B,C,D] f32
    int*         __restrict__ out_mask)// [B,C,S] i32
{
    __shared__ float xs[kS * kXStride];          // 57600 B
    __shared__ __align__(16) int ml[kS];         // masked labels (-1 = masked out)
    __shared__ float inv[kC];                    // 1/count (0 if empty)

    const int b   = blockIdx.x;
    const int tid = threadIdx.x;

    // ---- stage masked labels ----
    if (tid < kS) {
        const int lab = labels[b * kS + tid];
        const int m   = amask[b * kS + tid];
        ml[tid] = m ? lab : -1;
    }

    // ---- stage x[b] into LDS (float4 loads, padded-stride store) ----
    const float4* xg = (const float4*)(x + (size_t)b * kS * kD);
    for (int i = tid; i < (kS * kD / 4); i += 256) {
        const float4 v = xg[i];
        const int s = (i * 4) >> 6;   // /64
        const int d = (i * 4) & 63;
        float* p = &xs[s * kXStride + d];
        p[0] = v.x; p[1] = v.y; p[2] = v.z; p[3] = v.w;
    }
    __syncthreads();

    // ---- per-cluster counts -> reciprocals (thread == cluster) ----
    {
        const int c = tid;
        int cnt = 0;
        #pragma unroll 4
        for (int s = 0; s < kS; ++s) cnt += (ml[s] == c) ? 1 : 0;
        inv[c] = (cnt > 0) ? (1.0f / (float)cnt) : 0.0f;
    }
    __syncthreads();

    // ---- WMMA phase: D[c,d] = sum_s onehot[c,s] * x[s,d] ----
    // 16 c-tiles x 4 d-tiles = 64 tiles of 16x16; 8 waves -> 8 tiles each.
    const int wave = tid >> 5;       // wave32
    const int lane = tid & 31;
    const int half = lane >> 4;      // half-wave selects K pair / M+8
    const int lm   = lane & 15;

    float* out_emb_b = out_emb + (size_t)b * kC * kD;

    for (int t = wave; t < 64; t += 8) {
        const int c0 = (t >> 2) * 16;
        const int d0 = (t & 3) * 16;
        const int myc = c0 + lm;     // A-matrix row for this lane
        v8f acc = {};
        #pragma unroll 2
        for (int k = 0; k < kS; k += 4) {
            const int kk = k + 2 * half;
            const int l0 = ml[kk];
            const int l1 = ml[kk + 1];
            v2f a, bb;
            // A 16x4 f32 layout: lanes 0-15 hold K={k,k+1}, lanes 16-31 K={k+2,k+3}
            a[0] = (l0 == myc) ? 1.0f : 0.0f;
            a[1] = (l1 == myc) ? 1.0f : 0.0f;
            // B 4x16 f32: N across lanes, same half-wave K split
            bb[0] = xs[kk * kXStride + d0 + lm];
            bb[1] = xs[(kk + 1) * kXStride + d0 + lm];
            acc = __builtin_amdgcn_wmma_f32_16x16x4_f32(
                /*neg_a=*/false, a, /*neg_b=*/false, bb,
                /*c_mod=*/(short)0, acc, /*reuse_a=*/false, /*reuse_b=*/false);
        }
        // C/D layout: VGPR r -> row M = r + 8*half, col N = lm
        #pragma unroll
        for (int r = 0; r < 8; ++r) {
            const int c = c0 + r + 8 * half;
            out_emb_b[(size_t)c * kD + d0 + lm] = acc[r] * inv[c];
        }
    }

    // ---- int32 cluster_mask output, fully coalesced int4 stores ----
    int4* out_mask_b = (int4*)(out_mask + (size_t)b * kC * kS);
    // C*S/4 = 12800 int4 per batch; 256 threads -> 50 iterations
    for (int j = tid; j < (kC * kS / 4); j += 256) {
        const int c  = j / (kS / 4);          // j / 50
        const int s4 = (j - c * (kS / 4)) * 4;
        const int4 lv = *((const int4*)&ml[s4]);   // 16B-aligned LDS read
        int4 mv;
        mv.x = (lv.x == c) ? 1 : 0;
        mv.y = (lv.y == c) ? 1 : 0;
        mv.z = (lv.z == c) ? 1 : 0;
        mv.w = (lv.w == c) ? 1 : 0;
        out_mask_b[j] = mv;
    }
}

extern "C" void kernel_launch(void* const* d_in, const int* in_sizes, int n_in,
                              void* d_out, int out_size, void* d_ws, size_t ws_size,
                              hipStream_t stream) {
    const float* x      = (const float*)d_in[0];   // rq_item_embeddings [B,S,D] f32
    const int*   labels = (const int*)d_in[1];     // [B,S] i32
    const int*   amask  = (const int*)d_in[2];     // [B,S] i32

    float* out_emb  = (float*)d_out;                                   // [B,C,D] f32
    int*   out_mask = (int*)((char*)d_out +
                             (size_t)kB * kC * kD * sizeof(float));    // [B,C,S] i32

    hipLaunchKernelGGL(s3rec_cluster_mean_kernel, dim3(kB), dim3(256), 0, stream,
                       x, labels, amask, out_emb, out_mask);
}